// GCNConv_19868518711830
// MI455X (gfx1250) — compile-verified
//
#include <hip/hip_runtime.h>

typedef __attribute__((ext_vector_type(2))) float v2f;
typedef __attribute__((ext_vector_type(8))) float v8f;

#define N_NODES 16384
#define D 32
#define HASH_CAP (1u << 21)
#define HASH_MASK (HASH_CAP - 1u)

// -------------------------------------------------------------------------
// support = x @ weight  via V_WMMA_F32_16X16X4_F32.
// One wave32 per 16x16 output tile; K=32 consumed in 8 steps of 4.
// A 16x4 f32 layout (ISA 7.12.2): lane L holds M = L%16, K = 2*(L/16)+j (VGPR j).
// B 4x16 f32 mirrored: lane L holds N = L%16, K = 2*(L/16)+j.
// C/D 16x16 f32: VGPR v, lane L -> M = v + 8*(L/16), N = L%16.
// No divergence anywhere in this kernel (EXEC must be all ones for WMMA).
// -------------------------------------------------------------------------
__global__ void gcn_support_wmma(const float* __restrict__ x,
                                 const float* __restrict__ w,
                                 float* __restrict__ support) {
  int wave = (int)((blockIdx.x * blockDim.x + threadIdx.x) >> 5);
  int lane = (int)(threadIdx.x & 31);
  int tm = wave >> 1;   // 1024 M-tiles
  int tn = wave & 1;    // 2 N-tiles
  int m = lane & 15;
  int half = lane >> 4;

  const float* xrow = x + (tm * 16 + m) * D;
  v8f c = {};
#pragma unroll
  for (int k0 = 0; k0 < D; k0 += 4) {
    int k = k0 + 2 * half;
    v2f a, b;
    a.x = xrow[k + 0];
    a.y = xrow[k + 1];
    b.x = w[(k + 0) * D + tn * 16 + m];
    b.y = w[(k + 1) * D + tn * 16 + m];
    c = __builtin_amdgcn_wmma_f32_16x16x4_f32(false, a, false, b, (short)0, c,
                                              false, false);
  }
#pragma unroll
  for (int v = 0; v < 8; ++v) {
    support[(tm * 16 + v + 8 * half) * D + tn * 16 + m] = c[v];
  }
}

// -------------------------------------------------------------------------
// Re-initialize scratch every call (graph replays must be deterministic).
// -------------------------------------------------------------------------
__global__ void gcn_init(unsigned* __restrict__ table,
                         unsigned* __restrict__ deg,
                         float* __restrict__ accum) {
  unsigned tid = blockIdx.x * blockDim.x + threadIdx.x;
  if (tid < HASH_CAP) table[tid] = 0xFFFFFFFFu;
  if (tid < N_NODES) deg[tid] = 0u;
  if (tid < N_NODES * D) accum[tid] = 0.0f;
}

// -------------------------------------------------------------------------
// One wave32 per edge. Lane 0 inserts key (r<<14)|c into an open-addressing
// hash set (set semantics of adj.at[r,c].set(1.0)); if the edge is new, all
// 32 lanes atomically accumulate support[c,:] into accum[r,:] and lane 0
// bumps the distinct out-degree.
// -------------------------------------------------------------------------
__global__ void gcn_edges(const int* __restrict__ rows,
                          const int* __restrict__ cols,
                          const float* __restrict__ support,
                          unsigned* __restrict__ table,
                          unsigned* __restrict__ deg,
                          float* __restrict__ accum,
                          int num_edges) {
  int wave = (int)((blockIdx.x * blockDim.x + threadIdx.x) >> 5);
  int lane = (int)(threadIdx.x & 31);
  if (wave >= num_edges) return;
  int r = rows[wave];
  int c = cols[wave];
  int is_new = 0;
  if (lane == 0) {
    unsigned key = ((unsigned)r << 14) | (unsigned)c;   // r,c < 16384
    unsigned h = (key * 2654435761u) & HASH_MASK;
    for (;;) {
      unsigned old = atomicCAS(&table[h], 0xFFFFFFFFu, key);
      if (old == 0xFFFFFFFFu) {  // first time we see this (r,c)
        is_new = 1;
        atomicAdd(&deg[r], 1u);
        break;
      }
      if (old == key) break;     // duplicate edge: set semantics -> skip
      h = (h + 1u) & HASH_MASK;
    }
  }
  is_new = __shfl(is_new, 0, 32);
  if (is_new) {
    atomicAdd(&accum[r * D + lane], support[c * D + lane]);
  }
}

// -------------------------------------------------------------------------
// out[r,ch] = (accum[r,ch] + support[r,ch]) / (deg[r] + 1) + bias[ch]
// (self loop adds support[r] and +1 to the row sum on top of any existing
//  (r,r) edge, exactly matching adj + eye in the reference).
// -------------------------------------------------------------------------
__global__ void gcn_finalize(const float* __restrict__ accum,
                             const float* __restrict__ support,
                             const unsigned* __restrict__ deg,
                             const float* __restrict__ bias,
                             float* __restrict__ out) {
  int tid = (int)(blockIdx.x * blockDim.x + threadIdx.x);
  if (tid >= N_NODES * D) return;
  int r = tid >> 5;
  int ch = tid & 31;
  float denom = (float)deg[r] + 1.0f;
  out[tid] = (accum[tid] + support[tid]) / denom + bias[ch];
}

extern "C" void kernel_launch(void* const* d_in, const int* in_sizes, int n_in,
                              void* d_out, int out_size, void* d_ws, size_t ws_size,
                              hipStream_t stream) {
  const float* x = (const float*)d_in[0];
  const int* ei = (const int*)d_in[1];     // (2, E) int
  const float* w = (const float*)d_in[2];  // (32, 32)
  const float* bias = (const float*)d_in[3];
  float* out = (float*)d_out;
  int num_edges = in_sizes[1] / 2;

  // Workspace layout: hash table (8 MB) | deg (64 KB) | accum (2 MB) | support (2 MB)
  char* ws = (char*)d_ws;
  unsigned* table = (unsigned*)ws;
  unsigned* deg = (unsigned*)(ws + (size_t)HASH_CAP * 4);
  float* accum = (float*)(ws + (size_t)HASH_CAP * 4 + (size_t)N_NODES * 4);
  float* support = accum + (size_t)N_NODES * D;

  gcn_init<<<HASH_CAP / 256, 256, 0, stream>>>(table, deg, accum);

  // 2048 tiles, 8 waves (tiles) per 256-thread block -> 256 blocks
  gcn_support_wmma<<<(N_NODES / 16) * (D / 16) / 8, 256, 0, stream>>>(x, w, support);

  long long edge_threads = (long long)num_edges * 32;
  gcn_edges<<<(unsigned)((edge_threads + 255) / 256), 256, 0, stream>>>(
      ei, ei + num_edges, support, table, deg, accum, num_edges);

  gcn_finalize<<<(N_NODES * D + 255) / 256, 256, 0, stream>>>(accum, support,
                                                              deg, bias, out);
}